// DSSMs_66546223284831
// MI455X (gfx1250) — compile-verified
//
#include <hip/hip_runtime.h>
#include <hip/hip_bf16.h>

typedef __attribute__((ext_vector_type(2))) float v2f;
typedef __attribute__((ext_vector_type(8))) float v8f;

#define L_SEQ   8192
#define D_MODEL 256
#define D_INNER 512
#define D_STATE 16
#define DT_RANK 16
#define NCHUNK  128
#define TCHUNK  64    // NCHUNK * TCHUNK == L_SEQ

// ---------------------------------------------------------------------------
// fp32 WMMA GEMM:  C[M x N] = A[M x K] * B[N x K]^T  (+ epilogue)
// One wave per 16 x (16*NT) output tile: NT accumulators share one A fragment
// per k-step (1 A load + NT B loads -> NT v_wmma_f32_16x16x4_f32).
// EPI: 0 = none, 1 = softplus(acc + bias[col]), 2 = acc + bias[col] + addbuf
// ---------------------------------------------------------------------------
template<int NT, int EPI>
__global__ __launch_bounds__(32)
void wmma_gemm_kernel(const float* __restrict__ A, int lda,
                      const float* __restrict__ B, int ldb,
                      float* __restrict__ C, int ldc, int K,
                      const float* __restrict__ bias,
                      const float* __restrict__ addbuf)
{
    const int lane = threadIdx.x;
    const int nb = blockIdx.x;            // tile group along N (NT*16 wide)
    const int mb = blockIdx.y;            // 16-row tile along M
    const int rlo = lane & 15;            // M index for A, N index for B
    const int kh  = (lane >> 4) * 2;      // K sub-offset {0,2}

    const float* Arow = A + (long)(mb * 16 + rlo) * lda + kh;
    const float* Brow = B + (long)(nb * (16 * NT) + rlo) * ldb + kh;

    v8f acc[NT];
#pragma unroll
    for (int i = 0; i < NT; ++i) acc[i] = (v8f){0.f,0.f,0.f,0.f,0.f,0.f,0.f,0.f};

    for (int k = 0; k < K; k += 4) {
        __builtin_prefetch(Arow + k + 64, 0, 1);   // speculative; OOB dropped
        v2f a = *(const v2f*)(Arow + k);
        v2f b[NT];
#pragma unroll
        for (int i = 0; i < NT; ++i) {
            __builtin_prefetch(Brow + (long)i * 16 * ldb + k + 64, 0, 1);
            b[i] = *(const v2f*)(Brow + (long)i * 16 * ldb + k);
        }
#pragma unroll
        for (int i = 0; i < NT; ++i) {
            acc[i] = __builtin_amdgcn_wmma_f32_16x16x4_f32(
                false, a, false, b[i], (short)0, acc[i], false, false);
        }
    }

    const int rbase = mb * 16 + ((lane >> 4) * 8);
#pragma unroll
    for (int i = 0; i < NT; ++i) {
        const int col = nb * (16 * NT) + i * 16 + rlo;
#pragma unroll
        for (int r = 0; r < 8; ++r) {
            long idx = (long)(rbase + r) * ldc + col;
            float v = acc[i][r];
            if (EPI == 1) {
                v += bias[col];
                v = (v > 20.f) ? v : __logf(1.f + __expf(v));   // softplus
            } else if (EPI == 2) {
                v += bias[col] + addbuf[idx];
            }
            C[idx] = v;
        }
    }
}

// ---------------------------------------------------------------------------
// Depthwise causal conv (D_CONV=4) + SiLU.  x lives in xz[:, 0:512].
// ---------------------------------------------------------------------------
__global__ __launch_bounds__(256)
void conv_silu_kernel(const float* __restrict__ xz,
                      const float* __restrict__ conv_w,
                      const float* __restrict__ conv_b,
                      float* __restrict__ xc)
{
    int idx = blockIdx.x * 256 + threadIdx.x;   // t*512 + d
    int t = idx >> 9;
    int d = idx & 511;
    float acc = conv_b[d];
#pragma unroll
    for (int k = 0; k < 4; ++k) {
        int tt = t - 3 + k;
        float xv = (tt >= 0) ? xz[(long)tt * 1024 + d] : 0.f;
        acc = fmaf(conv_w[d * 4 + k], xv, acc);
    }
    xc[idx] = acc * (1.f / (1.f + __expf(-acc)));   // SiLU
}

// ---------------------------------------------------------------------------
// Chunked selective scan (linear diagonal recurrence => chunk-parallel).
// FIRST pass : h0 = 0, accumulate per-chunk transition product P and tail q.
// SECOND pass: h0 = h_in[chunk], emit ys[t,d] = sum_s h[s]*C_t[s].
// Lane == channel; 16 states in registers. B_t / C_t are wave-uniform loads.
// ---------------------------------------------------------------------------
template<bool FIRST>
__global__ __launch_bounds__(32)
void scan_chunk_kernel(const float* __restrict__ delta,
                       const float* __restrict__ xc,
                       const float* __restrict__ x_dbl,
                       const float* __restrict__ A_log,
                       const float* __restrict__ h_in,
                       float* __restrict__ Pout,
                       float* __restrict__ qout,
                       float* __restrict__ ys)
{
    const int lane  = threadIdx.x;
    const int chunk = blockIdx.x;
    const int d     = blockIdx.y * 32 + lane;

    float a[16], h[16], P[16];
#pragma unroll
    for (int s = 0; s < 16; ++s) a[s] = -__expf(A_log[d * 16 + s]);

    const long sbase = ((long)chunk * D_INNER + d) * 16;
    if (FIRST) {
#pragma unroll
        for (int s = 0; s < 16; ++s) { h[s] = 0.f; P[s] = 1.f; }
    } else {
#pragma unroll
        for (int i = 0; i < 4; ++i) {
            float4 hv = ((const float4*)(h_in + sbase))[i];
            h[4 * i + 0] = hv.x; h[4 * i + 1] = hv.y;
            h[4 * i + 2] = hv.z; h[4 * i + 3] = hv.w;
        }
    }

    const int t0 = chunk * TCHUNK;
    for (int t = t0; t < t0 + TCHUNK; ++t) {
        float dl = delta[(long)t * D_INNER + d];
        float xv = xc[(long)t * D_INNER + d];
        float dx = dl * xv;

        const float4* bp = (const float4*)(x_dbl + (long)t * 48 + DT_RANK);
        float Bv[16], Cv[16];
#pragma unroll
        for (int i = 0; i < 4; ++i) {
            float4 v = bp[i];                       // B_t (uniform)
            Bv[4 * i + 0] = v.x; Bv[4 * i + 1] = v.y;
            Bv[4 * i + 2] = v.z; Bv[4 * i + 3] = v.w;
        }
        if (!FIRST) {
#pragma unroll
            for (int i = 0; i < 4; ++i) {
                float4 v = bp[4 + i];               // C_t (uniform)
                Cv[4 * i + 0] = v.x; Cv[4 * i + 1] = v.y;
                Cv[4 * i + 2] = v.z; Cv[4 * i + 3] = v.w;
            }
        }

        float yacc = 0.f;
#pragma unroll
        for (int s = 0; s < 16; ++s) {
            float e = __expf(dl * a[s]);
            h[s] = fmaf(h[s], e, dx * Bv[s]);
            if (FIRST) P[s] *= e;
            else       yacc = fmaf(h[s], Cv[s], yacc);
        }
        if (!FIRST) ys[(long)t * D_INNER + d] = yacc;
    }

    if (FIRST) {
#pragma unroll
        for (int i = 0; i < 4; ++i) {
            ((float4*)(Pout + sbase))[i] =
                make_float4(P[4 * i], P[4 * i + 1], P[4 * i + 2], P[4 * i + 3]);
            ((float4*)(qout + sbase))[i] =
                make_float4(h[4 * i], h[4 * i + 1], h[4 * i + 2], h[4 * i + 3]);
        }
    }
}

// Sequential cross-chunk combine: h_in[c] = carry; carry = P[c]*carry + q[c].
__global__ __launch_bounds__(32)
void scan_combine_kernel(const float* __restrict__ P,
                         const float* __restrict__ q,
                         float* __restrict__ h_in)
{
    const int d = blockIdx.x * 32 + threadIdx.x;
    float carry[16];
#pragma unroll
    for (int s = 0; s < 16; ++s) carry[s] = 0.f;

    for (int c = 0; c < NCHUNK; ++c) {
        const long base = ((long)c * D_INNER + d) * 16;
#pragma unroll
        for (int i = 0; i < 4; ++i)
            ((float4*)(h_in + base))[i] = make_float4(
                carry[4 * i], carry[4 * i + 1], carry[4 * i + 2], carry[4 * i + 3]);
#pragma unroll
        for (int i = 0; i < 4; ++i) {
            float4 pv = ((const float4*)(P + base))[i];
            float4 qv = ((const float4*)(q + base))[i];
            carry[4 * i + 0] = fmaf(pv.x, carry[4 * i + 0], qv.x);
            carry[4 * i + 1] = fmaf(pv.y, carry[4 * i + 1], qv.y);
            carry[4 * i + 2] = fmaf(pv.z, carry[4 * i + 2], qv.z);
            carry[4 * i + 3] = fmaf(pv.w, carry[4 * i + 3], qv.w);
        }
    }
}

// y = (ys + D*xc) * silu(z); z = xz[:, 512:]. In-place into ys buffer.
__global__ __launch_bounds__(256)
void ygate_kernel(const float* __restrict__ xz,
                  const float* __restrict__ xc,
                  const float* __restrict__ Dvec,
                  float* __restrict__ ysy)
{
    int idx = blockIdx.x * 256 + threadIdx.x;   // t*512 + d
    int t = idx >> 9;
    int d = idx & 511;
    float z = xz[(long)t * 1024 + 512 + d];
    float sz = z * (1.f / (1.f + __expf(-z)));
    ysy[idx] = (ysy[idx] + Dvec[d] * xc[idx]) * sz;
}

// ---------------------------------------------------------------------------
// Fused double LayerNorm over rows of 256 (one block per row, 8 waves).
// ---------------------------------------------------------------------------
__device__ __forceinline__ float block_sum_256(float v, float* sred)
{
#pragma unroll
    for (int off = 16; off >= 1; off >>= 1) v += __shfl_xor(v, off, 32);
    int w = threadIdx.x >> 5;
    __syncthreads();
    if ((threadIdx.x & 31) == 0) sred[w] = v;
    __syncthreads();
    float r = sred[0];
#pragma unroll
    for (int i = 1; i < 8; ++i) r += sred[i];
    return r;
}

__global__ __launch_bounds__(256)
void double_layernorm_kernel(const float* __restrict__ in,
                             const float* __restrict__ g2, const float* __restrict__ b2,
                             const float* __restrict__ g1, const float* __restrict__ b1,
                             float* __restrict__ out)
{
    __shared__ float sred[8];
    const int row = blockIdx.x;
    const int tid = threadIdx.x;
    const float inv_n = 1.f / 256.f;

    float x = in[(long)row * 256 + tid];
    float mu  = block_sum_256(x, sred) * inv_n;
    float dx  = x - mu;
    float var = block_sum_256(dx * dx, sred) * inv_n;
    float v1  = dx * rsqrtf(var + 1e-5f) * g2[tid] + b2[tid];

    mu  = block_sum_256(v1, sred) * inv_n;
    dx  = v1 - mu;
    var = block_sum_256(dx * dx, sred) * inv_n;
    out[(long)row * 256 + tid] = dx * rsqrtf(var + 1e-5f) * g1[tid] + b1[tid];
}

// ---------------------------------------------------------------------------
extern "C" void kernel_launch(void* const* d_in, const int* in_sizes, int n_in,
                              void* d_out, int out_size, void* d_ws, size_t ws_size,
                              hipStream_t stream)
{
    const float* feat       = (const float*)d_in[0];   // [8192,256]
    const float* in_proj_w  = (const float*)d_in[1];   // [1024,256]
    const float* conv_w     = (const float*)d_in[2];   // [512,4]
    const float* conv_b     = (const float*)d_in[3];   // [512]
    const float* x_proj_w   = (const float*)d_in[4];   // [48,512]
    const float* dt_proj_w  = (const float*)d_in[5];   // [512,16]
    const float* dt_proj_b  = (const float*)d_in[6];   // [512]
    const float* A_log      = (const float*)d_in[7];   // [512,16]
    const float* Dvec       = (const float*)d_in[8];   // [512]
    const float* out_proj_w = (const float*)d_in[9];   // [256,512]
    const float* res_w      = (const float*)d_in[10];  // [256,256]
    const float* res_b      = (const float*)d_in[11];  // [256]
    const float* norm2_g    = (const float*)d_in[12];
    const float* norm2_b    = (const float*)d_in[13];
    const float* norm_g     = (const float*)d_in[14];
    const float* norm_b     = (const float*)d_in[15];
    float* outp = (float*)d_out;

    // Workspace partition (floats)
    float* ws = (float*)d_ws;
    float* xz    = ws;                         // 8192*1024
    float* xc    = xz    + (long)L_SEQ * 1024; // 8192*512
    float* x_dbl = xc    + (long)L_SEQ * 512;  // 8192*48
    float* delta = x_dbl + (long)L_SEQ * 48;   // 8192*512
    float* ysy   = delta + (long)L_SEQ * 512;  // 8192*512 (ys then gated y)
    float* Pbuf  = ysy   + (long)L_SEQ * 512;  // 128*512*16
    float* qbuf  = Pbuf  + (long)NCHUNK * D_INNER * 16;
    float* hbuf  = qbuf  + (long)NCHUNK * D_INNER * 16;
    float* mamba = hbuf  + (long)NCHUNK * D_INNER * 16;  // 8192*256

    dim3 wave(32);

    // 1) xz = feat @ in_proj_w^T   (M=8192, N=1024, K=256)
    wmma_gemm_kernel<4, 0><<<dim3(1024 / 64, L_SEQ / 16), wave, 0, stream>>>(
        feat, 256, in_proj_w, 256, xz, 1024, 256, nullptr, nullptr);

    // 2) depthwise conv + SiLU -> xc
    conv_silu_kernel<<<(L_SEQ * 512) / 256, 256, 0, stream>>>(xz, conv_w, conv_b, xc);

    // 3) x_dbl = xc @ x_proj_w^T   (N=48, K=512)
    wmma_gemm_kernel<3, 0><<<dim3(1, L_SEQ / 16), wave, 0, stream>>>(
        xc, 512, x_proj_w, 512, x_dbl, 48, 512, nullptr, nullptr);

    // 4) delta = softplus(dt @ dt_proj_w^T + b)   (A = x_dbl[:, :16], N=512, K=16)
    wmma_gemm_kernel<4, 1><<<dim3(512 / 64, L_SEQ / 16), wave, 0, stream>>>(
        x_dbl, 48, dt_proj_w, 16, delta, 512, 16, dt_proj_b, nullptr);

    // 5) chunked scan: pass 1 (P, q), combine, pass 2 (ys)
    scan_chunk_kernel<true><<<dim3(NCHUNK, D_INNER / 32), wave, 0, stream>>>(
        delta, xc, x_dbl, A_log, nullptr, Pbuf, qbuf, nullptr);
    scan_combine_kernel<<<D_INNER / 32, wave, 0, stream>>>(Pbuf, qbuf, hbuf);
    scan_chunk_kernel<false><<<dim3(NCHUNK, D_INNER / 32), wave, 0, stream>>>(
        delta, xc, x_dbl, A_log, hbuf, nullptr, nullptr, ysy);

    // 6) gate: y = (ys + D*xc) * silu(z)
    ygate_kernel<<<(L_SEQ * 512) / 256, 256, 0, stream>>>(xz, xc, Dvec, ysy);

    // 7) mamba_out = y @ out_proj_w^T   (N=256, K=512)
    wmma_gemm_kernel<4, 0><<<dim3(256 / 64, L_SEQ / 16), wave, 0, stream>>>(
        ysy, 512, out_proj_w, 512, mamba, 256, 512, nullptr, nullptr);

    // 8) mamba += feat @ res_w^T + res_b   (N=256, K=256, in-place add)
    wmma_gemm_kernel<4, 2><<<dim3(256 / 64, L_SEQ / 16), wave, 0, stream>>>(
        feat, 256, res_w, 256, mamba, 256, 256, res_b, mamba);

    // 9) double layernorm -> d_out
    double_layernorm_kernel<<<L_SEQ, 256, 0, stream>>>(
        mamba, norm2_g, norm2_b, norm_g, norm_b, outp);
}